// BoxLoss_46780783788225
// MI455X (gfx1250) — compile-verified
//
#include <hip/hip_runtime.h>
#include <hip/hip_bf16.h>

#define NB_B   16
#define NB_A   100000
#define NB_GT  32
#define THRESH 0.3f

typedef float v2f __attribute__((ext_vector_type(2)));
typedef float v8f __attribute__((ext_vector_type(8)));

__device__ __forceinline__ unsigned int fkey(float x) {
    unsigned int u = __float_as_uint(x);
    return (u & 0x80000000u) ? ~u : (u | 0x80000000u);
}

__device__ __forceinline__ float softplus0(float x) {
    // bce_with_logits(x, 0) = max(x,0) + log1p(exp(-|x|))
    return fmaxf(x, 0.f) + log1pf(expf(-fabsf(x)));
}

__device__ __forceinline__ float smooth_l1(float d) {
    float ad = fabsf(d);
    return ad < 1.f ? 0.5f * d * d : ad - 0.5f;
}

__device__ __forceinline__ float iou_one(float ax0, float ay0, float ax1, float ay1,
                                         float area_a,
                                         float gx0, float gy0, float gx1, float gy1) {
    float lx = fmaxf(ax0, gx0), ly = fmaxf(ay0, gy0);
    float rx = fminf(ax1, gx1), ry = fminf(ay1, gy1);
    float w = fmaxf(rx - lx, 0.f), h = fmaxf(ry - ly, 0.f);
    float inter = w * h;
    float area_g = (gx1 - gx0) * (gy1 - gy0);
    return inter / (area_a + area_g - inter);
}

// Block-wide f32 sum. Funnels 64 partials through one V_WMMA_F32_16X16X4_F32:
// A = 64 partials (16x4), B = ones => D rows are row-sums; total = sum of rows.
// Result valid on threads 0..31. `red` must hold NT+64 floats.
template <int NT>
__device__ float block_reduce_wmma(float v, float* red) {
    int tid = threadIdx.x;
    red[tid] = v;
    __syncthreads();
    float* r64 = red + NT;
    if (tid < 64) {
        float s = 0.f;
        #pragma unroll
        for (int i = 0; i < NT / 64; ++i) s += red[tid * (NT / 64) + i];
        r64[tid] = s;
    }
    __syncthreads();
    float total = 0.f;
    if (tid < 32) {                         // wave 0 only: uniform branch, EXEC all-ones
        v2f a; a.x = r64[2 * tid]; a.y = r64[2 * tid + 1];
        v2f bb; bb.x = 1.f; bb.y = 1.f;
        v8f c = {};
        c = __builtin_amdgcn_wmma_f32_16x16x4_f32(
                false, a, false, bb, (short)0, c, false, false);
        float t = c[0] + c[1] + c[2] + c[3] + c[4] + c[5] + c[6] + c[7];
        t += __shfl_xor(t, 16, 32);         // rows 0-7 half + rows 8-15 half
        total = t;
    }
    __syncthreads();
    return total;
}

// ---------------- K0: init workspace ----------------
__global__ void k_init(unsigned long long* best, unsigned int* n_pos, float* accum) {
    int t = blockIdx.x * 256 + threadIdx.x;
    if (t < NB_B * NB_GT) best[t] = 0ull;
    if (t < NB_B) n_pos[t] = 0u;
    if (t < 3) accum[t] = 0.f;
}

// ---------------- K1: best anchor per GT ----------------
__global__ void k_best(const float* __restrict__ anchors, const float* __restrict__ gts,
                       unsigned long long* __restrict__ best) {
    int b = blockIdx.y;
    int a = blockIdx.x * 256 + threadIdx.x;
    int tid = threadIdx.x;
    __shared__ float g4[NB_GT * 4];
    __shared__ unsigned long long lb[NB_GT];
    if (tid < NB_GT * 4) g4[tid] = gts[b * NB_GT * 4 + tid];
    if (tid < NB_GT) lb[tid] = 0ull;
    __syncthreads();
    if (a < NB_A) {
        float4 an = ((const float4*)anchors)[a];
        float area_a = (an.z - an.x) * (an.w - an.y);
        #pragma unroll 4
        for (int g = 0; g < NB_GT; ++g) {
            float i = iou_one(an.x, an.y, an.z, an.w, area_a,
                              g4[g * 4 + 0], g4[g * 4 + 1], g4[g * 4 + 2], g4[g * 4 + 3]);
            // pack: high = iou bits (iou>=0 so monotone), low = inverted index
            // => max key == max iou, ties pick lowest anchor index (argmax semantics)
            unsigned long long key =
                ((unsigned long long)__float_as_uint(i) << 32) |
                (unsigned long long)(0x7FFFFFFFu - (unsigned int)a);
            atomicMax(&lb[g], key);
        }
    }
    __syncthreads();
    if (tid < NB_GT) atomicMax(&best[b * NB_GT + tid], lb[tid]);
}

// ---------------- K2: match + box loss + pos BCE + n_pos + pos mask ----------------
__global__ void k_match(const float* __restrict__ labels, const float* __restrict__ offsets,
                        const float* __restrict__ anchors, const float* __restrict__ gts,
                        const unsigned long long* __restrict__ best,
                        unsigned char* __restrict__ posmask,
                        unsigned int* __restrict__ n_pos, float* __restrict__ accum) {
    int b = blockIdx.y;
    int a = blockIdx.x * 256 + threadIdx.x;
    int tid = threadIdx.x;
    __shared__ float g4[NB_GT * 4];
    __shared__ unsigned int bestA[NB_GT];
    __shared__ float red[256 + 64];
    __shared__ unsigned int sh_np;
    if (tid < NB_GT * 4) g4[tid] = gts[b * NB_GT * 4 + tid];
    if (tid < NB_GT)
        bestA[tid] = 0x7FFFFFFFu - (unsigned int)(best[b * NB_GT + tid] & 0xFFFFFFFFull);
    if (tid == 0) sh_np = 0u;
    __syncthreads();

    float localBox = 0.f, localPos = 0.f;
    int isPos = 0;
    if (a < NB_A) {
        size_t ba = (size_t)b * NB_A + a;
        __builtin_prefetch(labels + ba, 0, 0);            // global_prefetch_b8
        __builtin_prefetch(offsets + ba * 4, 0, 0);
        float4 an = ((const float4*)anchors)[a];
        float area_a = (an.z - an.x) * (an.w - an.y);
        float best_iou = -1.f; int best_g = 0;
        #pragma unroll 4
        for (int g = 0; g < NB_GT; ++g) {
            float i = iou_one(an.x, an.y, an.z, an.w, area_a,
                              g4[g * 4 + 0], g4[g * 4 + 1], g4[g * 4 + 2], g4[g * 4 + 3]);
            if (i > best_iou) { best_iou = i; best_g = g; }   // first max -> argmax
        }
        #pragma unroll
        for (int g = 0; g < NB_GT; ++g)
            if (bestA[g] == (unsigned int)a) { best_g = g; best_iou = 1.0f; }
        isPos = best_iou > THRESH;
        posmask[ba] = (unsigned char)isPos;
        if (isPos) {
            float gx0 = g4[best_g * 4 + 0], gy0 = g4[best_g * 4 + 1];
            float gx1 = g4[best_g * 4 + 2], gy1 = g4[best_g * 4 + 3];
            float pcx = (an.x + an.z) * 0.5f, pcy = (an.y + an.w) * 0.5f;
            float pw = an.z - an.x, ph = an.w - an.y;
            float gcx = (gx0 + gx1) * 0.5f, gcy = (gy0 + gy1) * 0.5f;
            float gw = gx1 - gx0, gh = gy1 - gy0;
            float t0 = (gcx - pcx) / (pw / 10.f);
            float t1 = (gcy - pcy) / (ph / 10.f);
            float t2 = logf(gw / pw) * 5.f;
            float t3 = logf(gh / ph) * 5.f;
            float4 po = ((const float4*)offsets)[ba];
            localBox = smooth_l1(po.x - t0) + smooth_l1(po.y - t1) +
                       smooth_l1(po.z - t2) + smooth_l1(po.w - t3);
            float x = labels[ba];
            localPos = fmaxf(x, 0.f) - x + log1pf(expf(-fabsf(x)));   // bce(x, 1)
        }
    }
    if (isPos) atomicAdd(&sh_np, 1u);
    float boxTot = block_reduce_wmma<256>(localBox, red);
    float posTot = block_reduce_wmma<256>(localPos, red);
    if (tid == 0) {
        if (boxTot != 0.f) atomicAdd(&accum[0], boxTot);
        if (posTot != 0.f) atomicAdd(&accum[1], posTot);
        if (sh_np) atomicAdd(&n_pos[b], sh_np);
    }
}

// ---------------- K3: per-row exact top-k (radix select) of negative softplus ----------------
__global__ void __launch_bounds__(1024)
k_topk(const float* __restrict__ labels, const unsigned char* __restrict__ posmask,
       const unsigned int* __restrict__ n_pos, float* __restrict__ accum) {
    int b = blockIdx.x;
    int tid = threadIdx.x;
    const float* row = labels + (size_t)b * NB_A;
    const unsigned char* pm = posmask + (size_t)b * NB_A;

    __shared__ unsigned int hist[256];
    __shared__ float red[1024 + 64];
    __shared__ unsigned int sh_prefix, sh_mask, sh_kk;
    __shared__ int sh_mode;   // 0 = select, 1 = sum-all negatives, 2 = nothing
    __shared__ float sh_xT;

    if (tid == 0) {
        unsigned int np = n_pos[b];
        unsigned long long k = 3ull * (unsigned long long)np;
        unsigned int C = NB_A - np;                 // negative candidate count
        if (np == 0u)                sh_mode = 2;
        else if (k >= (unsigned long long)C) sh_mode = 1;
        else                         sh_mode = 0;
        sh_prefix = 0u; sh_mask = 0u;
        sh_kk = (k >= (unsigned long long)C) ? C : (unsigned int)k;
    }
    __syncthreads();
    if (sh_mode == 2) return;                       // uniform early-out

    for (int r = 0; r < 4 && sh_mode == 0; ++r) {
        int shift = 24 - 8 * r;
        if (tid < 256) hist[tid] = 0u;
        __syncthreads();
        unsigned int pfx = sh_prefix, msk = sh_mask;
        for (int a = tid; a < NB_A; a += 1024) {
            if (!pm[a]) {
                unsigned int key = fkey(row[a]);
                if ((key & msk) == pfx)
                    atomicAdd(&hist[(key >> shift) & 255u], 1u);
            }
        }
        __syncthreads();
        if (tid == 0) {
            unsigned int kk = sh_kk, cum = 0u; int d = 0;
            for (int i = 255; i >= 0; --i) {
                unsigned int c = hist[i];
                if (cum + c >= kk) { d = i; sh_kk = kk - cum; break; }
                cum += c;
            }
            sh_prefix = pfx | ((unsigned int)d << shift);
            sh_mask = msk | (255u << shift);
        }
        __syncthreads();
    }

    unsigned int T = (sh_mode == 1) ? 0u : sh_prefix;  // all real-float keys > 0
    float local = 0.f;
    for (int a = tid; a < NB_A; a += 1024) {
        if (!pm[a]) {
            float x = row[a];
            unsigned int key = fkey(x);
            if (key > T) local += softplus0(x);
            else if (key == T && sh_mode == 0) sh_xT = x;  // benign same-value race
        }
    }
    __syncthreads();
    float tot = block_reduce_wmma<1024>(local, red);
    if (tid == 0) {
        float h = tot;
        if (sh_mode == 0 && sh_kk > 0u) h += (float)sh_kk * softplus0(sh_xT);
        atomicAdd(&accum[2], h);
    }
}

// ---------------- K4: finalize ----------------
__global__ void k_final(const unsigned int* __restrict__ n_pos,
                        const float* __restrict__ accum, float* __restrict__ out) {
    if (threadIdx.x == 0 && blockIdx.x == 0) {
        float npt = 0.f;
        for (int b = 0; b < NB_B; ++b) npt += (float)n_pos[b];
        float box = accum[0] / (npt * 4.f);
        float cls = (accum[2] + accum[1]) / npt;
        out[0] = box + cls;
        out[1] = box;
        out[2] = cls;
    }
}

extern "C" void kernel_launch(void* const* d_in, const int* in_sizes, int n_in,
                              void* d_out, int out_size, void* d_ws, size_t ws_size,
                              hipStream_t stream) {
    const float* labels  = (const float*)d_in[0];   // [B, A, 1]
    const float* offsets = (const float*)d_in[1];   // [B, A, 4]
    const float* anchors = (const float*)d_in[2];   // [A, 4]
    const float* gts     = (const float*)d_in[3];   // [B, NGT, 4]

    char* ws = (char*)d_ws;
    unsigned long long* best = (unsigned long long*)ws;                 // 4 KB
    unsigned char* posmask   = (unsigned char*)(ws + 4096);             // B*A bytes
    unsigned int* n_pos      = (unsigned int*)(ws + 4096 + (size_t)NB_B * NB_A);
    float* accum             = (float*)(ws + 4096 + (size_t)NB_B * NB_A + 64);

    dim3 grid2((NB_A + 255) / 256, NB_B);

    k_init <<<2, 256, 0, stream>>>(best, n_pos, accum);
    k_best <<<grid2, 256, 0, stream>>>(anchors, gts, best);
    k_match<<<grid2, 256, 0, stream>>>(labels, offsets, anchors, gts, best,
                                       posmask, n_pos, accum);
    k_topk <<<NB_B, 1024, 0, stream>>>(labels, posmask, n_pos, accum);
    k_final<<<1, 64, 0, stream>>>(n_pos, accum, (float*)d_out);
}